// PLIF_85401129713967
// MI455X (gfx1250) — compile-verified
//
#include <hip/hip_runtime.h>
#include <stdint.h>

// PLIF spiking recurrence, memory-bound: 64 MiB in + 64 MiB out @ 23.3 TB/s -> ~5.8 us floor.
// Load stream driven by the Tensor Data Mover (tensor_load_to_lds, quad-buffered LDS,
// up to 3 tiles in flight per wave => ~12 MB demand traffic in flight across 1024 waves),
// store stream via non-temporal b128 vector stores -> loads and stores on separate engines.

typedef __attribute__((ext_vector_type(4))) float    float4v;
typedef __attribute__((ext_vector_type(4))) uint32_t uint4v;
typedef __attribute__((ext_vector_type(8))) uint32_t uint8v;

#define PF    8                  // timesteps per TDM tile
#define BUFS  4                  // LDS buffers (BUFS-1 = 3 tiles in flight)
#define WAVE  32                 // wave32; one wave per workgroup
#define ROWF  (WAVE * 4)         // 128 floats per tile row (one float4 per lane)
#define ROWB  (ROWF * 4)         // 512 bytes per tile row

// ---------------------------------------------------------------------------
// TDM-driven kernel: grid = (N/ROWF, B), block = 32 threads (1 wave).
// ---------------------------------------------------------------------------
__global__ __launch_bounds__(WAVE) void plif_tdm_kernel(
    const float* __restrict__ x_seq,     // [T, B, N]
    const float* __restrict__ v0,        // [B, N]
    const float* __restrict__ decay,     // [N]
    const float* __restrict__ threshold, // [N]
    float* __restrict__ out,             // [T, B, N]
    int BN, int N, int T)
{
    __shared__ float4v lds_buf[BUFS * PF * WAVE];   // 16 KB quad buffer

    const int lane     = threadIdx.x;            // 0..31
    const int colChunk = blockIdx.x;             // chunk of ROWF floats within N
    const int batch    = blockIdx.y;
    const int colOff   = batch * N + colChunk * ROWF;  // float offset inside one timestep
    const int i4       = (colOff >> 2) + lane;         // this lane's float4 column
    const int cols4    = BN >> 2;

    // Per-neuron params: position within N only (2D grid -> no modulo).
    const int n4 = ((colChunk * ROWF) >> 2) + lane;
    const float4v d  = ((const float4v*)decay)[n4];
    const float4v th = ((const float4v*)threshold)[n4];
    float4v v = ((const float4v*)v0)[i4];

    // LDS byte offset of the buffer = low 32 bits of the flat address (ISA aperture rule).
    const uint32_t lds_base = (uint32_t)(uintptr_t)(&lds_buf[0]);
    const uint64_t gbase    = (uint64_t)(uintptr_t)x_seq + (uint64_t)colOff * 4ull;

    const int C = (T + PF - 1) / PF;             // number of PF-row chunks

    // Issue one TDM load: tile [rows x ROWF floats], row stride BN, into buffer chunk%BUFS.
    auto issue_tdm = [&](int chunk, int rows) {
        const uint64_t gaddr = gbase + (uint64_t)chunk * PF * (uint64_t)BN * 4ull;
        const uint32_t laddr = lds_base + (uint32_t)(chunk & (BUFS - 1)) * (PF * ROWB);

        uint4v g0;
        g0[0] = 1u;                                   // count=1, user descriptor
        g0[1] = laddr;                                // lds_addr (bytes)
        g0[2] = (uint32_t)gaddr;                      // global_addr[31:0]
        g0[3] = (uint32_t)(gaddr >> 32) | (2u << 30); // global_addr[56:32] | type=2

        uint8v g1;
        g1[0] = 2u << 16;                 // workgroup_mask=0 | data_size=2 (4 bytes)
        g1[1] = (uint32_t)ROWF << 16;     // tensor_dim0[15:0]  = ROWF
        g1[2] = (uint32_t)rows << 16;     // tensor_dim0 hi=0 | tensor_dim1[15:0]=rows
        g1[3] = (uint32_t)ROWF << 16;     // tensor_dim1 hi=0 | tile_dim0=ROWF
        g1[4] = (uint32_t)rows;           // tile_dim1=rows | tile_dim2=0
        g1[5] = (uint32_t)BN;             // tensor_dim0_stride[31:0] = BN elements
        g1[6] = 0u;                       // stride hi | tensor_dim1_stride (unused, 2D)
        g1[7] = 0u;

        asm volatile("tensor_load_to_lds %0, %1" :: "s"(g0), "s"(g1) : "memory");
    };

    // Prime the pipeline: chunks 0 .. BUFS-2 in flight before first consume.
#pragma unroll
    for (int p = 0; p < BUFS - 1; ++p) {
        if (p < C) {
            int rows = T - p * PF;
            if (rows > PF) rows = PF;
            issue_tdm(p, rows);
        }
    }

    for (int c = 0; c < C; ++c) {
        // Keep BUFS-1 chunks of DMA in flight while we consume chunk c.
        const int nc = c + BUFS - 1;
        if (nc < C) {
            int rows = T - nc * PF;
            if (rows > PF) rows = PF;
            issue_tdm(nc, rows);
        }

        // Guarantee chunk c complete: wait until at most `ahead` newer chunks outstanding.
        int ahead = C - 1 - c;
        if (ahead > BUFS - 1) ahead = BUFS - 1;
        if      (ahead >= 3) asm volatile("s_wait_tensorcnt 0x3" ::: "memory");
        else if (ahead == 2) asm volatile("s_wait_tensorcnt 0x2" ::: "memory");
        else if (ahead == 1) asm volatile("s_wait_tensorcnt 0x1" ::: "memory");
        else                 asm volatile("s_wait_tensorcnt 0x0" ::: "memory");

        const float4v* __restrict__ buf = &lds_buf[(c & (BUFS - 1)) * (PF * WAVE)];

#pragma unroll
        for (int k = 0; k < PF; ++k) {
            const int tt = c * PF + k;
            if (tt < T) {
                const float4v x = buf[k * WAVE + lane];   // ds_load_b128, conflict-free

                float4v s;
#pragma unroll
                for (int e = 0; e < 4; ++e) {
                    const float vv = __builtin_fmaf(d[e], v[e], x[e]);
                    const bool  sp = (vv >= th[e]);
                    s[e] = sp ? 1.0f : 0.0f;
                    v[e] = sp ? 0.0f : vv;
                }

                // Output never re-read: non-temporal b128 store on the vector pipe.
                __builtin_nontemporal_store(s, &((float4v*)out)[tt * cols4 + i4]);
            }
        }
    }
}

// ---------------------------------------------------------------------------
// Fallback: register-pipelined streaming kernel (any shape).
// ---------------------------------------------------------------------------
#define RPF 4
__global__ __launch_bounds__(256) void plif_scan_kernel(
    const float* __restrict__ x_seq, const float* __restrict__ v0,
    const float* __restrict__ decay, const float* __restrict__ threshold,
    float* __restrict__ out, int BN, int N, int T)
{
    const int i4 = blockIdx.x * blockDim.x + threadIdx.x;
    const int cols4 = BN >> 2;
    if (i4 >= cols4) return;

    const float4v* __restrict__ xp = (const float4v*)x_seq;
    float4v* __restrict__ op       = (float4v*)out;

    const int n4 = i4 % (N >> 2);
    const float4v d  = ((const float4v*)decay)[n4];
    const float4v th = ((const float4v*)threshold)[n4];
    float4v v = ((const float4v*)v0)[i4];

    float4v buf[RPF];
#pragma unroll
    for (int k = 0; k < RPF; ++k)
        if (k < T) buf[k] = xp[k * cols4 + i4];

    for (int t = 0; t < T; t += RPF) {
#pragma unroll
        for (int k = 0; k < RPF; ++k) {
            const int tt = t + k;
            if (tt < T) {
                const float4v x = buf[k];
                const int tp = tt + RPF;
                if (tp < T) buf[k] = xp[tp * cols4 + i4];

                float4v s;
#pragma unroll
                for (int e = 0; e < 4; ++e) {
                    const float vv = __builtin_fmaf(d[e], v[e], x[e]);
                    const bool  sp = (vv >= th[e]);
                    s[e] = sp ? 1.0f : 0.0f;
                    v[e] = sp ? 0.0f : vv;
                }
                __builtin_nontemporal_store(s, &op[tt * cols4 + i4]);
            }
        }
    }
}

// ---------------------------------------------------------------------------
extern "C" void kernel_launch(void* const* d_in, const int* in_sizes, int n_in,
                              void* d_out, int out_size, void* d_ws, size_t ws_size,
                              hipStream_t stream) {
    const float* x_seq     = (const float*)d_in[0];
    const float* v0        = (const float*)d_in[1];
    const float* decay     = (const float*)d_in[2];
    const float* threshold = (const float*)d_in[3];
    float* out             = (float*)d_out;

    const int N  = in_sizes[2];
    const int BN = in_sizes[1];
    const int T  = in_sizes[0] / BN;

    if ((N % ROWF) == 0 && (BN % N) == 0) {
        const int B = BN / N;
        dim3 grid(N / ROWF, B);                 // (32, 32) -> 1024 one-wave workgroups
        plif_tdm_kernel<<<grid, WAVE, 0, stream>>>(x_seq, v0, decay, threshold, out,
                                                   BN, N, T);
    } else {
        const int cols4 = BN / 4;
        const int block = 256;
        const int grid  = (cols4 + block - 1) / block;
        plif_scan_kernel<<<grid, block, 0, stream>>>(x_seq, v0, decay, threshold, out,
                                                     BN, N, T);
    }
}